// RINet_22462678958689
// MI455X (gfx1250) — compile-verified
//
#include <hip/hip_runtime.h>

typedef float v2f __attribute__((ext_vector_type(2)));
typedef float v4f __attribute__((ext_vector_type(4)));
typedef float v8f __attribute__((ext_vector_type(8)));
typedef int   v4i __attribute__((ext_vector_type(4)));

typedef __attribute__((address_space(1))) v4i* gv4i_p;  // global v4i*
typedef __attribute__((address_space(3))) v4i* lv4i_p;  // LDS v4i* (32-bit offset)

#define BS_    8192
#define N_     512
#define M_     1024
#define NITER_ 25

#define SLAB_K   64                     // K-rows of Gram A per LDS slab
#define LSTRIDE  528                    // 512 + 16 pad: rows K vs K+2 land 32 banks apart
#define SLABF    (SLAB_K * LSTRIDE)     // 33792 floats per buffer
#define LDS_BYTES (2 * SLABF * 4)       // 270336 B (<= 320 KB/WGP)

#if defined(__has_builtin)
#if __has_builtin(__builtin_amdgcn_global_load_async_to_lds_b128)
#define HAVE_ASYNC_BUILTIN 1
#endif
#endif

#if defined(__has_builtin)
#if __has_builtin(__builtin_amdgcn_s_wait_asynccnt)
#define WAIT_ASYNC(n) __builtin_amdgcn_s_wait_asynccnt(n)
#endif
#endif
#ifndef WAIT_ASYNC
#define WAIT_ASYNC(n) asm volatile("s_wait_asynccnt %0" ::"i"(n) : "memory")
#endif

static __device__ __forceinline__ v8f wmma_f32_k4(v2f a, v2f b, v8f c) {
  // D = A(16x4 f32) * B(4x16 f32) + C(16x16 f32)
  return __builtin_amdgcn_wmma_f32_16x16x4_f32(false, a, false, b, (short)0, c,
                                               false, false);
}

// Async 16B global -> LDS. lds_byte_off is the byte offset inside the (only)
// dynamic-LDS block (which starts at LDS offset 0); AS3 pointers are plain
// 32-bit offsets, AS1 pointers are the 64-bit VA, so both can be built from
// integers without addrspacecasts.
static __device__ __forceinline__ void cp16_async(const float* gsrc,
                                                  unsigned lds_byte_off) {
#if defined(HAVE_ASYNC_BUILTIN)
  __builtin_amdgcn_global_load_async_to_lds_b128(
      (gv4i_p)(unsigned long long)(size_t)gsrc, (lv4i_p)lds_byte_off, 0, 0);
#else
  // VDST = LDS byte offset VGPR, VADDR = 64-bit global address (GV mode)
  asm volatile("global_load_async_to_lds_b128 %0, %1, off" ::"v"(lds_byte_off),
               "v"(gsrc)
               : "memory");
#endif
}

// Cooperative stage of one 64x512 f32 K-slab of Gram A into LDS (padded rows).
// 256 threads x 16 B x 32 rounds = 128 KB.  32 async ops in flight per wave.
static __device__ __forceinline__ void stage_slab(const float* __restrict__ Ag,
                                                  int k0, unsigned buf_f,
                                                  int tid) {
#pragma unroll
  for (int it = 0; it < 32; ++it) {
    const int f4  = it * 256 + tid;  // float4 index within 64x128
    const int row = f4 >> 7;
    const int c4  = f4 & 127;
    cp16_async(Ag + (size_t)(k0 + row) * N_ + c4 * 4,
               (buf_f + (unsigned)(row * LSTRIDE + c4 * 4)) * 4u);
  }
}

// ---------------------------------------------------------------------------
// Kernel 1: yMF = y @ H^T, traj[0] = 0, s0 = yMF * 1/diag(A) (into s_final buf)
// Wave computes 16x64: A-op (M,K)=y[M][K], B-op (K,N)=H[N][K], both K-contig.
// ---------------------------------------------------------------------------
__global__ __launch_bounds__(128, 1) void ymf_kernel(
    const float* __restrict__ y, const float* __restrict__ H,
    const float* __restrict__ Ag, float* __restrict__ yMF,
    float* __restrict__ traj0, float* __restrict__ s0) {
  const int lane = threadIdx.x & 31;
  const int wid  = blockIdx.x * 4 + (threadIdx.x >> 5);
  const int m0   = (wid >> 3) * 16;
  const int n0   = (wid & 7) * 64;
  const int half = lane >> 4;
  const int l15  = lane & 15;

  v8f acc[4] = {};
  const float* yrow = y + (size_t)(m0 + l15) * M_ + 2 * half;
  const float* hrow[4];
#pragma unroll
  for (int j = 0; j < 4; ++j)
    hrow[j] = H + (size_t)(n0 + 16 * j + l15) * M_ + 2 * half;

  for (int k = 0; k < M_; k += 4) {
    const v2f a = *(const v2f*)(yrow + k);
#pragma unroll
    for (int j = 0; j < 4; ++j) {
      const v2f b = *(const v2f*)(hrow[j] + k);
      acc[j] = wmma_f32_k4(a, b, acc[j]);
    }
  }

#pragma unroll
  for (int j = 0; j < 4; ++j) {
    const int n = n0 + 16 * j + l15;
    const float dv = 1.0f / Ag[(size_t)n * N_ + n];
#pragma unroll
    for (int r = 0; r < 8; ++r) {
      const size_t o = (size_t)(m0 + r + 8 * half) * N_ + n;
      yMF[o]   = acc[j][r];
      traj0[o] = 0.0f;
      s0[o]    = acc[j][r] * dv;
    }
  }
}

// ---------------------------------------------------------------------------
// Kernel 2: 25 fused Richardson iterations, stateless per iteration.
// WG = 256 thr = 8 waves = 2 M-tiles x 4 N-groups over 32 rows.
// prev state comes from traj[t-1] (L2-hot, same-WG producer/consumer).
// Gram A streamed K-slab-wise into double-buffered LDS via async loads.
// ---------------------------------------------------------------------------
__global__ __launch_bounds__(256, 1) void iter_kernel(
    const float* __restrict__ yMF, const float* __restrict__ Ag,
    const float* __restrict__ invw, float* __restrict__ traj,
    float* __restrict__ s0_sfinal) {
  extern __shared__ float slds[];
  const int tid  = threadIdx.x;
  const int lane = tid & 31;
  const int wv   = tid >> 5;   // 0..7
  const int mi   = wv >> 2;    // M-tile in WG
  const int g    = wv & 3;     // N-group of 128 cols
  const int half = lane >> 4;
  const int l15  = lane & 15;
  const int m0   = blockIdx.x * 32 + mi * 16;
  const float w  = invw[0];

  for (int t = 1; t <= NITER_; ++t) {
    const float* prev =
        (t == 1) ? s0_sfinal : traj + (size_t)(t - 1) * BS_ * N_;
    float* trajT = traj + (size_t)t * BS_ * N_;
    const float* aprow = prev + (size_t)(m0 + l15) * N_ + 2 * half;

    stage_slab(Ag, 0, 0u, tid);  // prologue fill of buffer 0
    v8f acc[8] = {};

#pragma unroll 1
    for (int si = 0; si < 8; ++si) {
      if (si < 7)  // overlap next slab's fill with this slab's WMMAs
        stage_slab(Ag, (si + 1) * SLAB_K, (unsigned)(((si + 1) & 1) * SLABF),
                   tid);
      if (si < 7) {
        WAIT_ASYNC(32);  // in-order: oldest 32 (= slab si) have landed
      } else {
        WAIT_ASYNC(0);
      }
      __syncthreads();  // all 8 waves' fill portions visible
      const float* cur = slds + (si & 1) * SLABF;

#pragma unroll 2
      for (int kk = 0; kk < SLAB_K; kk += 4) {
        // A-op: lane l -> (M = l&15, K = si*64 + kk + 2*(l>>4) + {0,1})
        const v2f a = *(const v2f*)(aprow + si * SLAB_K + kk);
        const float* brow = cur + (size_t)(kk + 2 * half) * LSTRIDE;
#pragma unroll
        for (int jj = 0; jj < 8; ++jj) {
          const int n = g * 128 + jj * 16 + l15;
          v2f b;
          b.x = brow[n];            // row K   of Gram A (from LDS)
          b.y = brow[LSTRIDE + n];  // row K+1
          acc[jj] = wmma_f32_k4(a, b, acc[jj]);
        }
      }
      __syncthreads();  // everyone done reading `cur` before it is refilled
    }

    // s_t = s_{t-1} + w * (yMF - s_{t-1} @ A); emit traj[t] (and s_final)
#pragma unroll
    for (int jj = 0; jj < 8; ++jj) {
      const int n = g * 128 + jj * 16 + l15;
#pragma unroll
      for (int r = 0; r < 8; ++r) {
        const size_t o = (size_t)(m0 + r + 8 * half) * N_ + n;
        const float sn = prev[o] + w * (yMF[o] - acc[jj][r]);
        trajT[o] = sn;
        if (t == NITER_) s0_sfinal[o] = sn;
      }
    }
    // next iteration's first __syncthreads orders these stores vs. A-op loads
  }
}

// ---------------------------------------------------------------------------
extern "C" void kernel_launch(void* const* d_in, const int* in_sizes, int n_in,
                              void* d_out, int out_size, void* d_ws,
                              size_t ws_size, hipStream_t stream) {
  (void)in_sizes; (void)n_in; (void)out_size; (void)ws_size;
  const float* y  = (const float*)d_in[0];  // [8192,1024]
  const float* H  = (const float*)d_in[1];  // [512,1024]
  const float* Ag = (const float*)d_in[2];  // [512,512]
  const float* iw = (const float*)d_in[3];  // [1]
  // d_in[4] = num_itr (fixed 25)

  float* out     = (float*)d_out;
  float* s_final = out;                     // doubles as s0 buffer
  float* traj    = out + (size_t)BS_ * N_;  // [26,8192,512]
  float* yMF     = (float*)d_ws;            // [8192,512] scratch

  ymf_kernel<<<dim3((BS_ / 16) * (N_ / 64) / 4), dim3(128), 0, stream>>>(
      y, H, Ag, yMF, traj /* traj[0]=0 */, s_final /* s0 */);

  iter_kernel<<<dim3(BS_ / 32), dim3(256), LDS_BYTES, stream>>>(yMF, Ag, iw,
                                                                traj, s_final);
}